// RegionTargetPt_74062416053518
// MI455X (gfx1250) — compile-verified
//
#include <hip/hip_runtime.h>
#include <stdint.h>

// Problem constants (match reference setup_inputs)
#define B_  32
#define A_  5
#define H_  52
#define W_  52
#define T_  50
#define HW_ (H_ * W_)                 // 2704
#define CELLS_PER_B (A_ * HW_)        // 13520
#define NCELL (B_ * CELLS_PER_B)      // 432640
#define S10_ (2 * A_ * HW_)           // 27040 per-batch stride of 10-ch tensors
#define N10_ (B_ * S10_)              // 865280 elements per 10-ch tensor
#define S5_  CELLS_PER_B              // 13520
#define N5_  NCELL                    // 432640
#define POS_THRESH_ 0.6f

// ---------------------------------------------------------------------------
// Kernel 0: reset winner map to -1 (d_ws is not re-initialized by harness)
// ---------------------------------------------------------------------------
__global__ void fill_winner(int* __restrict__ winner) {
    int idx = blockIdx.x * blockDim.x + threadIdx.x;
    if (idx < NCELL) winner[idx] = -1;
}

// ---------------------------------------------------------------------------
// Kernel 1: per-(b,t) anchor argmax + scatter-max of gt index into its cell
// ---------------------------------------------------------------------------
__global__ void assign_kernel(const float* __restrict__ truth,
                              const float* __restrict__ biases,
                              int* __restrict__ winner) {
    int idx = blockIdx.x * blockDim.x + threadIdx.x;
    if (idx >= B_ * T_) return;
    int b = idx / T_;
    int t = idx - b * T_;

    const float* tr = truth + (b * T_ + t) * 5;
    float tx = tr[0], ty = tr[1], tw = tr[2], th = tr[3];

    // argmax over anchors of centered shape-IoU (first max wins, like argmax)
    float best = -1.0f;
    int n = 0;
#pragma unroll
    for (int a = 0; a < A_; ++a) {
        float bw0 = biases[2 * a]     * (1.0f / W_);
        float bh0 = biases[2 * a + 1] * (1.0f / H_);
        // (ir2-il2) == min(bw0, tw), (ib2-it2) == min(bh0, th)
        float iw = fmaxf(fminf(bw0, tw), 0.0f);
        float ih = fmaxf(fminf(bh0, th), 0.0f);
        float ov = iw * ih;
        float iou = ov / (bw0 * bh0 + tw * th - ov);
        if (iou > best) { best = iou; n = a; }
    }

    int ti = (int)(tx * W_);      // trunc-toward-zero, same as astype(int32)
    if (ti >= W_) ti = W_;        // reference clamps to W (still invalid)
    int tj = (int)(ty * H_);
    if (tj >= H_) tj = H_;

    bool invalid_pos = (tx <= 0.0f) || (tx >= 1.0f) || (ty <= 0.0f) || (ty >= 1.0f);
    bool valid = !invalid_pos && (ti >= 0) && (tj >= 0) &&
                 (ti < W_) && (tj < H_) && (tw > 0.0f) && (th > 0.0f);
    if (valid) {
        int cell = ((b * A_ + n) * H_ + tj) * W_ + ti;
        atomicMax(&winner[cell], t);   // global_atomic_max_i32
    }
}

// ---------------------------------------------------------------------------
// Kernel 2: main per-cell pass. Block = 256 threads (8 wave32), grid (53, B).
// Truth table for batch b (250 floats) staged into LDS with the CDNA5 async
// global->LDS path, drained via s_wait_asynccnt, then reused T_ times per
// thread as broadcast ds_load_b32 reads.
// ---------------------------------------------------------------------------
__global__ void region_main(const float* __restrict__ xy,
                            const float* __restrict__ wh,
                            const float* __restrict__ obj,
                            const float* __restrict__ truth,
                            const float* __restrict__ biases,
                            const int* __restrict__ winner,
                            float* __restrict__ out) {
    __shared__ float s_truth[T_ * 5];
    __shared__ float s_tl[T_], s_tr[T_], s_tt[T_], s_tb[T_], s_ta[T_];

    const int b = blockIdx.y;
    const int c = blockIdx.x * blockDim.x + threadIdx.x;

    // ---- async stage truth[b] (250 dwords) into LDS --------------------
    {
        int lane = threadIdx.x;
        int idx  = (lane < T_ * 5) ? lane : (T_ * 5 - 1);  // tail lanes dup last elem
        unsigned lds_addr = (unsigned)(uintptr_t)(&s_truth[idx]);
        unsigned goff     = (unsigned)((b * T_ * 5 + idx) * sizeof(float));
        asm volatile("global_load_async_to_lds_b32 %0, %1, %2"
                     :
                     : "v"(lds_addr), "v"(goff), "s"(truth)
                     : "memory");
        asm volatile("s_wait_asynccnt 0" ::: "memory");
    }
    __syncthreads();

    // ---- precompute per-truth corners + area in LDS --------------------
    if (threadIdx.x < T_) {
        int t = threadIdx.x;
        float tx = s_truth[5 * t + 0];
        float ty = s_truth[5 * t + 1];
        float tw = s_truth[5 * t + 2];
        float th = s_truth[5 * t + 3];
        s_tl[t] = tx - 0.5f * tw;
        s_tr[t] = tx + 0.5f * tw;
        s_tt[t] = ty - 0.5f * th;
        s_tb[t] = ty + 0.5f * th;
        s_ta[t] = tw * th;
    }
    __syncthreads();

    if (c >= CELLS_PER_B) return;

    const int a   = c / HW_;
    const int rem = c - a * HW_;
    const int j   = rem / W_;
    const int i   = rem - j * W_;

    const int baseXY = (b * 2 * A_ + a) * HW_ + rem;   // (B, 2A, H, W)
    const float xv = xy[baseXY];
    const float yv = xy[baseXY + A_ * HW_];
    const float wv = wh[baseXY];
    const float hv = wh[baseXY + A_ * HW_];
    const float o  = obj[(b * A_ + a) * HW_ + rem];

    const float bia_w = biases[2 * a];
    const float bia_h = biases[2 * a + 1];

    const float bx = (xv + (float)i) * (1.0f / W_);
    const float by = (yv + (float)j) * (1.0f / H_);
    const float bw = __expf(wv) * bia_w * (1.0f / W_);
    const float bh = __expf(hv) * bia_h * (1.0f / H_);
    const float bl = bx - 0.5f * bw, br = bx + 0.5f * bw;
    const float bt = by - 0.5f * bh, bb = by + 0.5f * bh;
    const float barea = bw * bh;

    const int w     = winner[b * CELLS_PER_B + c];
    const int wsafe = (w < 0) ? 0 : w;
    const bool has  = (w >= 0);

    float maxiou = 0.0f;
    float iou_w  = 0.0f;
#pragma unroll 5
    for (int t = 0; t < T_; ++t) {
        float il  = fmaxf(bl, s_tl[t]);
        float ir  = fminf(br, s_tr[t]);
        float it_ = fmaxf(bt, s_tt[t]);
        float ib  = fminf(bb, s_tb[t]);
        float ov  = fmaxf(ir - il, 0.0f) * fmaxf(ib - it_, 0.0f);
        float iou = ov / (barea + s_ta[t] - ov);
        maxiou = fmaxf(maxiou, iou);
        iou_w  = (t == wsafe) ? iou : iou_w;
    }
    const bool ign = maxiou > POS_THRESH_;

    const float stx  = s_truth[5 * wsafe + 0];
    const float sty  = s_truth[5 * wsafe + 1];
    const float stw  = s_truth[5 * wsafe + 2];
    const float sth  = s_truth[5 * wsafe + 3];
    const float scls = s_truth[5 * wsafe + 4];

    const float new_tx = stx * (float)W_ - (float)i;
    const float new_ty = sty * (float)H_ - (float)j;
    const float new_tw = __logf(stw * (float)W_ / bia_w);
    const float new_th = __logf(sth * (float)H_ / bia_h);
    const float wgt    = has ? (2.0f - stw * sth) : 0.0f;   // COORD_SCALE = 1

    float* t_xy    = out;
    float* t_wh    = out + N10_;
    float* t_wgt   = out + 2 * N10_;
    float* t_obj   = out + 3 * N10_;
    float* t_noobj = out + 3 * N10_ + N5_;
    float* t_lab   = out + 3 * N10_ + 2 * N5_;

    const int o10 = b * S10_ + c;          // c == a*HW + rem
    const int o5  = b * S5_  + c;

    t_xy [o10]            = has ? new_tx : xv;
    t_xy [o10 + A_ * HW_] = has ? new_ty : yv;
    t_wh [o10]            = has ? new_tw : wv;
    t_wh [o10 + A_ * HW_] = has ? new_th : hv;
    t_wgt[o10]            = wgt;
    t_wgt[o10 + A_ * HW_] = wgt;
    t_obj  [o5] = has ? iou_w : o;
    t_noobj[o5] = (has || ign) ? o : 0.0f;
    t_lab  [o5] = has ? scls : -1.0f;
}

// ---------------------------------------------------------------------------
extern "C" void kernel_launch(void* const* d_in, const int* in_sizes, int n_in,
                              void* d_out, int out_size, void* d_ws, size_t ws_size,
                              hipStream_t stream) {
    const float* xy     = (const float*)d_in[0];
    const float* wh     = (const float*)d_in[1];
    const float* obj    = (const float*)d_in[2];
    const float* truth  = (const float*)d_in[3];
    const float* biases = (const float*)d_in[4];
    float* out  = (float*)d_out;
    int* winner = (int*)d_ws;   // NCELL ints = 1.73 MB scratch

    fill_winner<<<(NCELL + 255) / 256, 256, 0, stream>>>(winner);
    assign_kernel<<<(B_ * T_ + 255) / 256, 256, 0, stream>>>(truth, biases, winner);

    dim3 grid((CELLS_PER_B + 255) / 256, B_);
    region_main<<<grid, 256, 0, stream>>>(xy, wh, obj, truth, biases, winner, out);
}